// GCN_45071386804958
// MI455X (gfx1250) — compile-verified
//
#include <hip/hip_runtime.h>

typedef __attribute__((ext_vector_type(2))) float v2f;
typedef __attribute__((ext_vector_type(8))) float v8f;

static constexpr int GRAPHS = 512;

// ---------- degree / normalization ----------
__global__ void k_degree(const int* __restrict__ dst, float* __restrict__ deg, int E) {
  int e = blockIdx.x * blockDim.x + threadIdx.x;
  if (e < E) atomicAdd(&deg[dst[e]], 1.0f);
}

__global__ void k_dis(float* __restrict__ deg, int N) {
  int i = blockIdx.x * blockDim.x + threadIdx.x;
  if (i < N) deg[i] = rsqrtf(deg[i] + 1.0f);  // +1 = self loop; always > 0
}

// LDS byte-address of a __shared__ object (addrspace(3) pointers are 32-bit)
__device__ __forceinline__ unsigned lds_addr_of(const void* p) {
  return (unsigned)(uintptr_t)(__attribute__((address_space(3))) const void*)p;
}

// ---------- WMMA f32 GEMM: out[nrows x OUTF] = A[nrows x 128] @ W[128 x OUTF] ----------
// One wave32 per 16-row strip; W staged in LDS via gfx1250 async global->LDS copies;
// K stepped by 4 via V_WMMA_F32_16X16X4_F32 with a depth-1 software pipeline on A.
// A frag layout (ISA 32-bit A 16x4): lanes 0-15 -> K0,K1 ; lanes 16-31 -> K2,K3 (M = lane&15).
// C/D layout: VGPR r -> row r (lanes 0-15) / row r+8 (lanes 16-31), col = lane&15.
template<int OUTF, bool FUSE_RELU_BIAS>
__global__ __launch_bounds__(256) void k_gemm_wmma(
    const float* __restrict__ A, const float* __restrict__ W,
    const float* __restrict__ abias, float* __restrict__ out, int nrows) {
  __shared__ __align__(16) float sW[128 * OUTF];

  // Stage W with GLOBAL_LOAD_ASYNC_TO_LDS_B128 (ASYNCcnt path, no VGPR round trip).
  for (int i = threadIdx.x * 4; i < 128 * OUTF; i += 256 * 4) {
    unsigned laddr = lds_addr_of(&sW[i]);
    const float* gp = W + i;
    asm volatile("global_load_async_to_lds_b128 %0, %1, off"
                 :: "v"(laddr), "v"(gp) : "memory");
  }
  asm volatile("s_wait_asynccnt 0x0" ::: "memory");
  __syncthreads();

  const int lane   = threadIdx.x & 31;
  const int wave   = threadIdx.x >> 5;
  const int mtile  = blockIdx.x * 8 + wave;
  const int mtiles = nrows >> 4;            // nrows is a multiple of 16 (100000)
  if (mtile >= mtiles) return;              // wave-uniform: EXEC stays all-ones for WMMA

  const int l15   = lane & 15;
  const int khalf = lane >> 4;              // 0 -> K pair {0,1}, 1 -> K pair {2,3}

  const float* arow = A + (size_t)(mtile * 16 + l15) * 128;

  auto loadA = [&](int k) -> v2f {
    const int ka = k + 2 * khalf;
    float a0 = arow[ka];
    float a1 = arow[ka + 1];
    if (FUSE_RELU_BIAS) {                   // fused relu(h1 + b1) of previous layer
      a0 = fmaxf(a0 + abias[ka], 0.0f);
      a1 = fmaxf(a1 + abias[ka + 1], 0.0f);
    }
    v2f a; a.x = a0; a.y = a1;
    return a;
  };

  v8f acc[OUTF / 16] = {};
  v2f a_cur = loadA(0);

  for (int k = 0; k < 128; k += 4) {
    v2f a_nxt = a_cur;
    if (k + 4 < 128) a_nxt = loadA(k + 4);  // prefetch: hides global latency behind WMMAs
    const int ka = k + 2 * khalf;
#pragma unroll
    for (int nt = 0; nt < OUTF / 16; ++nt) {
      v2f b;
      b.x = sW[ka * OUTF + nt * 16 + l15];
      b.y = sW[(ka + 1) * OUTF + nt * 16 + l15];
      acc[nt] = __builtin_amdgcn_wmma_f32_16x16x4_f32(
          false, a_cur, false, b, (short)0, acc[nt], false, false);
    }
    a_cur = a_nxt;
  }

  float* orow = out + (size_t)(mtile * 16 + khalf * 8) * OUTF;
#pragma unroll
  for (int nt = 0; nt < OUTF / 16; ++nt) {
#pragma unroll
    for (int r = 0; r < 8; ++r) {
      orow[(size_t)r * OUTF + nt * 16 + l15] = acc[nt][r];
    }
  }
}

// ---------- edge aggregation: out[dst] += xw[src] * dis[src]*dis[dst] ----------
// One wave per edge (incl. N virtual self-loop edges); lane covers F/32 contiguous floats.
template<int F>
__global__ void k_aggregate(const float* __restrict__ xw, const int* __restrict__ src,
                            const int* __restrict__ dst, const float* __restrict__ dis,
                            float* __restrict__ out, int E, int N) {
  constexpr int VEC = F / 32;
  const int wv   = (blockIdx.x * blockDim.x + threadIdx.x) >> 5;
  const int lane = threadIdx.x & 31;
  if (wv >= E + N) return;
  int s, d;
  if (wv < E) { s = src[wv]; d = dst[wv]; }
  else        { s = d = wv - E; }           // self loop
  const float norm = dis[s] * dis[d];
  const float* xs = xw + (size_t)s * F + lane * VEC;
  float*       od = out + (size_t)d * F + lane * VEC;
#pragma unroll
  for (int i = 0; i < VEC; ++i) atomicAdd(&od[i], xs[i] * norm);
}

// ---------- bias+relu + graph pooling ----------
__global__ void k_pool(const float* __restrict__ h2, const float* __restrict__ b2,
                       const int* __restrict__ batch, float* __restrict__ pooled, int N) {
  int idx = blockIdx.x * blockDim.x + threadIdx.x;
  if (idx >= N * 64) return;
  int i = idx >> 6, f = idx & 63;
  float v = fmaxf(h2[idx] + b2[f], 0.0f);
  atomicAdd(&pooled[(size_t)batch[i] * 64 + f], v);
}

// ---------- batch-norm statistics (population var, matching jnp.var) ----------
__global__ void k_bn(const float* __restrict__ pooled, float* __restrict__ meanvar) {
  int f = threadIdx.x;                      // 64 threads, coalesced column reads
  float s = 0.0f, ss = 0.0f;
  for (int g = 0; g < GRAPHS; ++g) {
    float v = pooled[g * 64 + f];
    s += v; ss += v * v;
  }
  float mean = s * (1.0f / GRAPHS);
  float var  = ss * (1.0f / GRAPHS) - mean * mean;
  meanvar[f]      = mean;
  meanvar[64 + f] = rsqrtf(var + 1e-5f);
}

// ---------- BN affine + MLP head; also emit h = pooled ----------
__global__ void k_head(const float* __restrict__ pooled, const float* __restrict__ meanvar,
                       const float* __restrict__ gamma, const float* __restrict__ beta,
                       const float* __restrict__ Wo1, const float* __restrict__ bo1,
                       const float* __restrict__ Wo2, const float* __restrict__ bo2,
                       float* __restrict__ out) {
  __shared__ float xn[64];
  __shared__ float t[24];
  const int g = blockIdx.x;
  const int f = threadIdx.x;                // 64 threads
  float p = pooled[g * 64 + f];
  out[GRAPHS + g * 64 + f] = p;             // second output: h = stop_gradient(pooled)
  xn[f] = (p - meanvar[f]) * meanvar[64 + f] * gamma[f] + beta[f];
  __syncthreads();
  if (f < 24) {
    float acc = bo1[f];
    for (int k = 0; k < 64; ++k) acc += xn[k] * Wo1[k * 24 + f];
    t[f] = fmaxf(acc, 0.0f);
  }
  __syncthreads();
  if (f == 0) {
    float acc = bo2[0];
    for (int j = 0; j < 24; ++j) acc += t[j] * Wo2[j];
    out[g] = acc;
  }
}

extern "C" void kernel_launch(void* const* d_in, const int* in_sizes, int n_in,
                              void* d_out, int out_size, void* d_ws, size_t ws_size,
                              hipStream_t stream) {
  const float* x     = (const float*)d_in[0];
  const int*   ei    = (const int*)d_in[1];
  const int*   batch = (const int*)d_in[2];
  const float* W1    = (const float*)d_in[3];
  const float* b1    = (const float*)d_in[4];
  const float* W2    = (const float*)d_in[5];
  const float* b2    = (const float*)d_in[6];
  const float* gamma = (const float*)d_in[7];
  const float* beta  = (const float*)d_in[8];
  const float* Wo1   = (const float*)d_in[9];
  const float* bo1   = (const float*)d_in[10];
  const float* Wo2   = (const float*)d_in[11];
  const float* bo2   = (const float*)d_in[12];
  float* out = (float*)d_out;

  const int N = in_sizes[0] / 128;
  const int E = in_sizes[1] / 2;
  const int* srcp = ei;
  const int* dstp = ei + E;

  // workspace layout (floats): dis[N] | buf1[N*128] | buf2[N*128] | pooled[G*64] | meanvar[128]
  float* dis    = (float*)d_ws;
  float* buf1   = dis + N;
  float* buf2   = buf1 + (size_t)N * 128;
  float* pooled = buf2 + (size_t)N * 128;
  float* meanvar = pooled + GRAPHS * 64;

  const int mtiles  = N / 16;
  const int gemmGrd = (mtiles + 7) / 8;
  const int totE    = E + N;

  // normalization
  hipMemsetAsync(dis, 0, (size_t)N * sizeof(float), stream);
  k_degree<<<(E + 255) / 256, 256, 0, stream>>>(dstp, dis, E);
  k_dis<<<(N + 255) / 256, 256, 0, stream>>>(dis, N);

  // layer 1: xw1 = x @ W1 ; h1 = scatter(norm * xw1[src]) (bias+relu deferred into GEMM2)
  k_gemm_wmma<128, false><<<gemmGrd, 256, 0, stream>>>(x, W1, nullptr, buf1, N);
  hipMemsetAsync(buf2, 0, (size_t)N * 128 * sizeof(float), stream);
  k_aggregate<128><<<(totE + 7) / 8, 256, 0, stream>>>(buf1, srcp, dstp, dis, buf2, E, N);

  // layer 2: xw2 = relu(h1 + b1) @ W2 ; h2 = scatter(norm * xw2[src])
  k_gemm_wmma<64, true><<<gemmGrd, 256, 0, stream>>>(buf2, W2, b1, buf1, N);
  hipMemsetAsync(buf2, 0, (size_t)N * 64 * sizeof(float), stream);
  k_aggregate<64><<<(totE + 7) / 8, 256, 0, stream>>>(buf1, srcp, dstp, dis, buf2, E, N);

  // pooling + BN + head
  hipMemsetAsync(pooled, 0, GRAPHS * 64 * sizeof(float), stream);
  k_pool<<<(N * 64 + 255) / 256, 256, 0, stream>>>(buf2, b2, batch, pooled, N);
  k_bn<<<1, 64, 0, stream>>>(pooled, meanvar);
  k_head<<<GRAPHS, 64, 0, stream>>>(pooled, meanvar, gamma, beta, Wo1, bo1, Wo2, bo2, out);
}